// Attention_5111011082776
// MI455X (gfx1250) — compile-verified
//
#include <hip/hip_runtime.h>

typedef float v2f __attribute__((ext_vector_type(2)));
typedef float v8f __attribute__((ext_vector_type(8)));

// Problem constants
constexpr int B  = 32;
constexpr int H  = 32;
constexpr int KH = 8;
constexpr int D  = 128;
constexpr int L  = 4096;
constexpr int G  = H / KH;                       // 4 query rows per kv head
constexpr float SCALE = 0.08838834764831845f;

// Kernel config
constexpr int SPLIT      = 8;                    // flash-decode splits along L
constexpr int NW         = 4;                    // waves per workgroup
constexpr int NTHREADS   = NW * 32;
constexpr int CH         = 16;                   // positions per WMMA tile
constexpr int POS_PER_WG = L / SPLIT;            // 512
constexpr int CHUNKS     = POS_PER_WG / CH;      // 32
constexpr int CHPW       = CHUNKS / NW;          // 8 chunks per wave (contiguous)
constexpr int KSTRIDE    = 132;                  // padded LDS row stride (floats)
constexpr int TILE_F     = CH * KSTRIDE;         // 2112 floats per K/V tile
constexpr int WAVE_F     = 4 * TILE_F + 64;      // 2x(K+V) double buffer + P(4x16)
constexpr int QS_OFF     = NW * WAVE_F;          // Q staging, G rows * KSTRIDE
constexpr int COMB_OFF   = QS_OFF + G * KSTRIDE;
constexpr int COMB_PW    = G * 130;              // per-wave: [g][m,l,acc(128)]
constexpr int LDS_FLOATS = COMB_OFF + NW * COMB_PW;   // ~146.6 KB
constexpr float NEG_BIG  = -3.0e38f;

// Async global->LDS copy of 16B per lane (one 512B row per wave-instruction).
// Tracked by ASYNCcnt; completion is in-order for global async loads.
__device__ __forceinline__ void async_row_b128(uint32_t lds_byte, uint64_t gaddr)
{
    asm volatile("global_load_async_to_lds_b128 %0, %1, off"
                 :: "v"(lds_byte), "v"(gaddr) : "memory");
}

__global__ __launch_bounds__(NTHREADS, 1)
void attn_fd_kernel(const float* __restrict__ q,
                    const float* __restrict__ kfreshp,
                    const float* __restrict__ vfreshp,
                    const float* __restrict__ kc,
                    const float* __restrict__ vc,
                    const int*   __restrict__ idx,
                    float*       __restrict__ part)
{
    extern __shared__ float smem[];
    const int tid   = threadIdx.x;
    const int lane  = tid & 31;
    const int wave  = tid >> 5;
    const int bk    = blockIdx.x;            // b*KH + kh
    const int b     = bk / KH;
    const int kh    = bk % KH;
    const int split = blockIdx.y;
    const int half  = lane >> 4;             // 0: K rows 0-1, 1: K rows 2-3
    const int laneM = lane & 15;
    const v2f zero2 = {0.f, 0.f};

    // ---- stage Q (pre-scaled) into LDS: 4 rows x 128, padded stride ----
    {
        const int g  = tid >> 5;             // 0..3 (NW==G)
        const int d0 = (tid & 31) * 4;
        const float* qrow = q + ((size_t)(b * H + kh * G + g)) * D;
        smem[QS_OFF + g * KSTRIDE + d0 + 0] = qrow[d0 + 0] * SCALE;
        smem[QS_OFF + g * KSTRIDE + d0 + 1] = qrow[d0 + 1] * SCALE;
        smem[QS_OFF + g * KSTRIDE + d0 + 2] = qrow[d0 + 2] * SCALE;
        smem[QS_OFF + g * KSTRIDE + d0 + 3] = qrow[d0 + 3] * SCALE;
    }
    __syncthreads();

    const uint32_t smem_base = (uint32_t)(uintptr_t)smem;     // LDS byte offset
    const uint32_t wave_base = smem_base + (uint32_t)(wave * WAVE_F) * 4u;
    const uint32_t voff      = (uint32_t)(lane * 16);         // per-lane 16B slice
    float* const pt = smem + wave * WAVE_F + 4 * TILE_F;      // P tile [4][16]

    const int    last_row = idx[b * L + (L - 1)];
    const float* kfresh   = kfreshp + (size_t)(b * KH + kh) * D;
    const float* vfresh   = vfreshp + (size_t)(b * KH + kh) * D;
    const float* kc_h     = kc + (size_t)kh * D;              // + row*KH*D later
    const float* vc_h     = vc + (size_t)kh * D;

    // ---- preload this wave's 128 row indices (contiguous block) ----
    const int wave_pos = split * POS_PER_WG + wave * (CHPW * CH);
    const int4 idxv = ((const int4*)(idx + (size_t)b * L + wave_pos))[lane];

    // ---- preload Q A-operands for all 32 K-steps into registers ----
    v2f qa[32];
    {
        const float* qs = smem + QS_OFF;
        #pragma unroll
        for (int s = 0; s < 32; ++s) {
            const int k0 = 4 * s + 2 * half;
            v2f t = *(const v2f*)(qs + laneM * KSTRIDE + k0);
            qa[s] = (laneM < G) ? t : zero2;
        }
    }

    const v8f vzero = {0.f,0.f,0.f,0.f,0.f,0.f,0.f,0.f};
    float mrow[G], lrow[G];
    v8f acc[8];
    #pragma unroll
    for (int g = 0; g < G; ++g) { mrow[g] = NEG_BIG; lrow[g] = 0.f; }
    #pragma unroll
    for (int t = 0; t < 8; ++t) acc[t] = vzero;

    // ---- async-stage one chunk's K and V tiles into LDS buffer `buf` ----
    auto stage_chunk = [&](int n, int buf) {
        const uint32_t ktoff = wave_base + (uint32_t)(buf * 2 * TILE_F) * 4u + voff;
        const uint32_t vtoff = ktoff + (uint32_t)TILE_F * 4u;
        #pragma unroll
        for (int i = 0; i < CH; ++i) {
            const int src_lane = 4 * n + (i >> 2);
            int rv;
            if      ((i & 3) == 0) rv = idxv.x;
            else if ((i & 3) == 1) rv = idxv.y;
            else if ((i & 3) == 2) rv = idxv.z;
            else                   rv = idxv.w;
            const int row = __builtin_amdgcn_readlane(rv, src_lane);  // uniform
            const bool fresh = (row == last_row);
            const float* ks = fresh ? kfresh : (kc_h + (size_t)row * (KH * D));
            const float* vs = fresh ? vfresh : (vc_h + (size_t)row * (KH * D));
            async_row_b128(ktoff + (uint32_t)(i * KSTRIDE) * 4u,
                           (uint64_t)(uintptr_t)ks + voff);
            async_row_b128(vtoff + (uint32_t)(i * KSTRIDE) * 4u,
                           (uint64_t)(uintptr_t)vs + voff);
        }
    };

    stage_chunk(0, 0);

    for (int n = 0; n < CHPW; ++n) {
        const int buf = n & 1;
        if (n + 1 < CHPW) {
            stage_chunk(n + 1, buf ^ 1);                 // prefetch next chunk
            asm volatile("s_wait_asynccnt 0x20" ::: "memory");  // cur 32 done
        } else {
            asm volatile("s_wait_asynccnt 0x0" ::: "memory");
        }
        const float* kt = smem + wave * WAVE_F + buf * 2 * TILE_F;
        const float* vt = kt + TILE_F;

        // ---- scores = (SCALE*Q) @ K^T : M=16(pad 4), N=16 pos, K=4 x 32 ----
        v8f sc = vzero;
        #pragma unroll
        for (int s = 0; s < 32; ++s) {
            const int k0 = 4 * s + 2 * half;
            const v2f bb = *(const v2f*)(kt + laneM * KSTRIDE + k0);  // ds_load_b64
            sc = __builtin_amdgcn_wmma_f32_16x16x4_f32(false, qa[s], false, bb,
                                                       (short)0, sc, false, false);
        }

        // ---- online softmax per query row g (cols live in lanes 0-15) ----
        #pragma unroll
        for (int g = 0; g < G; ++g) {
            float x = (lane < 16) ? sc[g] : NEG_BIG;
            #pragma unroll
            for (int off = 16; off > 0; off >>= 1)
                x = fmaxf(x, __shfl_xor(x, off));
            const float mnew = fmaxf(mrow[g], x);
            const float pe   = (lane < 16) ? __expf(sc[g] - mnew) : 0.f;
            float rs = pe;
            #pragma unroll
            for (int off = 16; off > 0; off >>= 1)
                rs += __shfl_xor(rs, off);
            const float alpha = __expf(mrow[g] - mnew);
            lrow[g] = lrow[g] * alpha + rs;
            mrow[g] = mnew;
            if (lane < 16) pt[g * 16 + lane] = pe;
            #pragma unroll
            for (int t = 0; t < 8; ++t) acc[t][g] *= alpha;   // rows 8-15 stay 0
        }

        // ---- O += P @ V : 8 D-tiles (N=16), K=16 positions in 4 steps ----
        #pragma unroll
        for (int s = 0; s < 4; ++s) {
            const int kp = 4 * s + 2 * half;
            v2f pa = *(const v2f*)(pt + laneM * 16 + kp);      // ds_load_b64
            pa = (laneM < G) ? pa : zero2;
            #pragma unroll
            for (int t = 0; t < 8; ++t) {
                v2f bv;
                bv[0] = vt[(kp + 0) * KSTRIDE + t * 16 + laneM];
                bv[1] = vt[(kp + 1) * KSTRIDE + t * 16 + laneM];
                acc[t] = __builtin_amdgcn_wmma_f32_16x16x4_f32(false, pa, false, bv,
                                                               (short)0, acc[t],
                                                               false, false);
            }
        }
    }

    // ---- per-wave partials -> LDS ----
    float* cb = smem + COMB_OFF + wave * COMB_PW;
    if (lane < 16) {
        #pragma unroll
        for (int g = 0; g < G; ++g) {
            if (lane == 0) { cb[g * 130 + 0] = mrow[g]; cb[g * 130 + 1] = lrow[g]; }
            #pragma unroll
            for (int t = 0; t < 8; ++t)
                cb[g * 130 + 2 + t * 16 + lane] = acc[t][g];
        }
    }
    __syncthreads();

    // ---- combine waves, emit split partial: [bk][split][g][130] ----
    {
        const int g  = tid >> 5;             // 0..3
        const int d0 = (tid & 31) * 4;
        float M = NEG_BIG;
        #pragma unroll
        for (int w = 0; w < NW; ++w)
            M = fmaxf(M, smem[COMB_OFF + w * COMB_PW + g * 130]);
        float Ls = 0.f, o0 = 0.f, o1 = 0.f, o2 = 0.f, o3 = 0.f;
        #pragma unroll
        for (int w = 0; w < NW; ++w) {
            const float* cw = smem + COMB_OFF + w * COMB_PW + g * 130;
            const float sw = __expf(cw[0] - M);
            Ls += sw * cw[1];
            o0 += sw * cw[2 + d0 + 0];
            o1 += sw * cw[2 + d0 + 1];
            o2 += sw * cw[2 + d0 + 2];
            o3 += sw * cw[2 + d0 + 3];
        }
        float* pp = part + (((size_t)bk * SPLIT + split) * G + g) * 130;
        pp[2 + d0 + 0] = o0;
        pp[2 + d0 + 1] = o1;
        pp[2 + d0 + 2] = o2;
        pp[2 + d0 + 3] = o3;
        if (d0 == 0) { pp[0] = M; pp[1] = Ls; }
    }
}

__global__ __launch_bounds__(128, 1)
void attn_combine_kernel(const float* __restrict__ part, float* __restrict__ out)
{
    const int bk = blockIdx.x;               // b*KH + kh
    const int b  = bk / KH;
    const int kh = bk % KH;
    const int d  = threadIdx.x;              // 0..127
    #pragma unroll
    for (int g = 0; g < G; ++g) {
        const float* base = part + (((size_t)bk * SPLIT) * G + g) * 130;
        float M = NEG_BIG;
        #pragma unroll
        for (int s = 0; s < SPLIT; ++s)
            M = fmaxf(M, base[(size_t)s * G * 130 + 0]);
        float Ls = 0.f, o = 0.f;
        #pragma unroll
        for (int s = 0; s < SPLIT; ++s) {
            const float* ps = base + (size_t)s * G * 130;
            const float w = __expf(ps[0] - M);
            Ls += w * ps[1];
            o  += w * ps[2 + d];
        }
        out[(size_t)b * (H * D) + (kh * G + g) * D + d] = o / Ls;
    }
}

extern "C" void kernel_launch(void* const* d_in, const int* in_sizes, int n_in,
                              void* d_out, int out_size, void* d_ws, size_t ws_size,
                              hipStream_t stream)
{
    const float* q   = (const float*)d_in[0];
    const float* k   = (const float*)d_in[1];
    const float* v   = (const float*)d_in[2];
    const float* kc  = (const float*)d_in[3];
    const float* vc  = (const float*)d_in[4];
    const int*   idx = (const int*)d_in[5];
    float* part = (float*)d_ws;              // B*KH*SPLIT*G*130 floats ~= 4.3 MB
    float* out  = (float*)d_out;

    dim3 grid(B * KH, SPLIT);
    const size_t ldsBytes = (size_t)LDS_FLOATS * sizeof(float);
    attn_fd_kernel<<<grid, NTHREADS, ldsBytes, stream>>>(q, k, v, kc, vc, idx, part);
    attn_combine_kernel<<<B * KH, 128, 0, stream>>>(part, out);
}